// WeightedGraphConvolutionLayer_61615600828800
// MI455X (gfx1250) — compile-verified
//
#include <hip/hip_runtime.h>
#include <hip/hip_bf16.h>

// Problem constants (match reference)
#define BB   8
#define NN   20000
#define DIN  256
#define DOUT 256
#define EDG  640000

typedef float v2f __attribute__((ext_vector_type(2)));
typedef float v8f __attribute__((ext_vector_type(8)));

#define LDS_STRIDE 260  // 256 + 4 pad: per-lane row stride -> 4-bank step, conflict-free

// ---------------------------------------------------------------------------
// Kernel 1: x[b] = feats[b] @ W  using V_WMMA_F32_16X16X4_F32 (fp32 precision)
// Block = 256 threads (8 waves). Block computes a 16-row strip x full 256 cols.
// Each wave owns two 16x16 column tiles (wid*16 and (wid+8)*16), sharing the
// LDS-staged A tile.
// ---------------------------------------------------------------------------
__global__ __launch_bounds__(256) void wgc_gemm_kernel(
    const float* __restrict__ feats, const float* __restrict__ weight,
    float* __restrict__ x)
{
    __shared__ float As[16 * LDS_STRIDE];

    const int rowBase = blockIdx.x * 16;
    const int b       = blockIdx.y;
    const size_t fOff = (size_t)b * NN * DIN;
    const int tid     = threadIdx.x;

    // Cooperative, coalesced load of the 16x256 A tile into LDS.
    #pragma unroll
    for (int i = 0; i < 16; ++i) {
        As[i * LDS_STRIDE + tid] = feats[fOff + (size_t)(rowBase + i) * DIN + tid];
    }
    __syncthreads();

    const int wid  = tid >> 5;
    const int lane = tid & 31;
    const int m    = lane & 15;                // A row (M) within tile
    const int koff = (lane >> 4) << 1;         // lanes 16-31 hold K+2,K+3
    const int n0   = wid * 16 + (lane & 15);   // first column tile
    const int n1   = (wid + 8) * 16 + (lane & 15); // second column tile

    v8f c0 = {};
    v8f c1 = {};

    for (int k = 0; k < DIN; k += 4) {
        const int kk = k + koff;
        v2f a;
        a.x = As[m * LDS_STRIDE + kk];
        a.y = As[m * LDS_STRIDE + kk + 1];
        v2f b0;
        b0.x = weight[kk * DOUT + n0];
        b0.y = weight[(kk + 1) * DOUT + n0];
        v2f b1;
        b1.x = weight[kk * DOUT + n1];
        b1.y = weight[(kk + 1) * DOUT + n1];

        // 8 args: (neg_a, A, neg_b, B, c_mod, C, reuse_a, reuse_b)
        c0 = __builtin_amdgcn_wmma_f32_16x16x4_f32(false, a, false, b0,
                                                   (short)0, c0, false, false);
        c1 = __builtin_amdgcn_wmma_f32_16x16x4_f32(false, a, false, b1,
                                                   (short)0, c1, false, false);
    }

    // C/D layout: lane -> N, VGPR v -> M = v + (lane>=16 ? 8 : 0)
    const size_t xOff = (size_t)b * NN * DOUT;
    const int mrow = rowBase + ((lane >> 4) << 3);
    #pragma unroll
    for (int v = 0; v < 8; ++v) {
        x[xOff + (size_t)(mrow + v) * DOUT + n0] = c0[v];
        x[xOff + (size_t)(mrow + v) * DOUT + n1] = c1[v];
    }
}

// ---------------------------------------------------------------------------
// Kernel 2: seed out with broadcast bias (out = bias, scatter adds on top)
// One block per (b, n) row; thread t owns feature channel t.
// ---------------------------------------------------------------------------
__global__ __launch_bounds__(256) void wgc_bias_kernel(
    const float* __restrict__ bias, float* __restrict__ out)
{
    out[(size_t)blockIdx.x * DOUT + threadIdx.x] = bias[threadIdx.x];
}

// ---------------------------------------------------------------------------
// Kernel 3: per-edge scatter  out[b][row] += (wv[e]*adj[e]) * x[b][col]
// Block of 256 threads handles 16 edges; thread t owns channel t.
// Edge index loads are wave-uniform -> scalar loads. fp32 atomics hit L2.
// ---------------------------------------------------------------------------
#define EDGES_PER_BLOCK 16
__global__ __launch_bounds__(256) void wgc_scatter_kernel(
    const float* __restrict__ x, const float* __restrict__ wv,
    const float* __restrict__ av, const int* __restrict__ rows,
    const int* __restrict__ cols, float* __restrict__ out)
{
    const int t       = threadIdx.x;
    const int b       = blockIdx.y;
    const size_t off  = (size_t)b * NN * DOUT;
    const int eBase   = blockIdx.x * EDGES_PER_BLOCK;

    #pragma unroll
    for (int i = 0; i < EDGES_PER_BLOCK; ++i) {
        const int e = eBase + i;
        const int r = rows[e];
        const int c = cols[e];
        const float w = wv[e] * av[e];
        const float val = w * x[off + (size_t)c * DOUT + t];
        unsafeAtomicAdd(&out[off + (size_t)r * DOUT + t], val);
    }
}

// ---------------------------------------------------------------------------
extern "C" void kernel_launch(void* const* d_in, const int* in_sizes, int n_in,
                              void* d_out, int out_size, void* d_ws, size_t ws_size,
                              hipStream_t stream) {
    const float* feats  = (const float*)d_in[0];
    const float* weight = (const float*)d_in[1];
    const float* bias   = (const float*)d_in[2];
    const float* wvals  = (const float*)d_in[3];
    const float* avals  = (const float*)d_in[4];
    const int*   erows  = (const int*)d_in[5];
    const int*   ecols  = (const int*)d_in[6];
    float* out = (float*)d_out;
    float* x   = (float*)d_ws;  // B*N*DOUT fp32 intermediate (164 MB)

    // 1) Dense GEMM via WMMA: grid = (N/16 row tiles, B)
    wgc_gemm_kernel<<<dim3(NN / 16, BB), 256, 0, stream>>>(feats, weight, x);

    // 2) Initialize output with bias broadcast: one block per (b, n) row
    wgc_bias_kernel<<<BB * NN, 256, 0, stream>>>(bias, out);

    // 3) Edge scatter with fp32 atomics: grid = (E/16, B)
    wgc_scatter_kernel<<<dim3(EDG / EDGES_PER_BLOCK, BB), 256, 0, stream>>>(
        x, wvals, avals, erows, ecols, out);
}